// GraphAttention_78546361909516
// MI455X (gfx1250) — compile-verified
//
#include <hip/hip_runtime.h>
#include <math.h>

#define NROW 8192
#define NCOL 8192
#define FDIM 256
#define SLOPE 0.2f

typedef __attribute__((ext_vector_type(2))) float v2f;
typedef __attribute__((ext_vector_type(4))) float v4f;
typedef __attribute__((ext_vector_type(8))) float v8f;

// ---------------------------------------------------------------------------
// Kernel 1: e = h @ a  via V_WMMA_F32_16X16X4_F32.
// One wave (32 threads) per 16-row tile. B (4x16) is `a` padded to 16 cols
// with zeros; `a` is staged in LDS and the zero-padding is done with a
// 0/1 mask multiply (no divergent loads in the K-loop).
//
// f32 A-matrix 16x4 layout: lanes 0-15 -> M=0..15 ; VGPR0 holds K=0 (lanes
// 0-15) / K=2 (lanes 16-31), VGPR1 holds K=1 / K=3.  B 4x16 is symmetric
// (rows=K striped across VGPRs, cols=N across lanes).  C layout: VGPR r,
// lanes 0-15 -> M=r, N=lane ; lanes 16-31 -> M=8+r, N=lane-16.
// ---------------------------------------------------------------------------
__global__ __launch_bounds__(32)
void gat_e_wmma_kernel(const float* __restrict__ h,
                       const float* __restrict__ a,
                       float* __restrict__ e1,
                       float* __restrict__ e2)
{
    __shared__ float s_a[FDIM * 2];          // 2 KB: a is (256, 2) row-major

    const int row0 = blockIdx.x * 16;
    const int lane = threadIdx.x;

    // Stage `a` into LDS (32 lanes x 4 b128 loads).
    {
        const v4f* src = (const v4f*)a;
        v4f* dst = (v4f*)s_a;
        for (int j = lane; j < FDIM * 2 / 4; j += 32)
            dst[j] = src[j];
    }
    __syncthreads();

    const int half = lane >> 4;              // 0 -> K base +0 ; 1 -> K base +2
    const int l16  = lane & 15;              // A: row within tile ; B/C: column
    const float colmask = (l16 < 2) ? 1.0f : 0.0f;
    const int   col     = l16 & 1;           // safe column index for all lanes

    v8f c = {};
    const float* hrow = h + (size_t)(row0 + l16) * FDIM;

    for (int k0 = 0; k0 < FDIM; k0 += 4) {
        const int ka = k0 + 2 * half;
        v2f av, bv;
        av.x = hrow[ka];
        av.y = hrow[ka + 1];
        // B[k][n] = a[k][n] for n < 2, else 0  (mask-multiply, no divergence)
        bv.x = s_a[ka * 2 + col]       * colmask;
        bv.y = s_a[(ka + 1) * 2 + col] * colmask;
        c = __builtin_amdgcn_wmma_f32_16x16x4_f32(
                /*neg_a=*/false, av, /*neg_b=*/false, bv,
                /*c_mod=*/(short)0, c, /*reuse_a=*/false, /*reuse_b=*/false);
    }

    // Column 0 of C -> e1, column 1 -> e2.
    if (l16 < 2) {
        float* dst = (l16 == 0) ? e1 : e2;
        const int rbase = row0 + 8 * half;
#pragma unroll
        for (int r = 0; r < 8; ++r)
            dst[rbase + r] = c[r];
    }
}

// ---------------------------------------------------------------------------
// Kernel 2: masked leaky-ReLU softmax over each row, scores recomputed from
// e1[i] + e2[j] (e2 staged in LDS).  One block (256 thr, 8 waves) per row.
// Adjacency row = 8 KB, read 3x but L0/L2 resident.  Output via NT b128.
// ---------------------------------------------------------------------------
__device__ __forceinline__ float leaky(float s) {
    return (s >= 0.0f) ? s : SLOPE * s;
}

__global__ __launch_bounds__(256)
void gat_softmax_kernel(const unsigned char* __restrict__ adj,
                        const float* __restrict__ e1,
                        const float* __restrict__ e2,
                        float* __restrict__ out)
{
    __shared__ float s_e2[NCOL];     // 32 KB
    __shared__ float s_red[8];

    const int row = blockIdx.x;
    const int tid = threadIdx.x;

    // Stage e2 into LDS with b128 loads.
    {
        const v4f* src = (const v4f*)e2;
        v4f* dst = (v4f*)s_e2;
        for (int j = tid; j < NCOL / 4; j += 256)
            dst[j] = src[j];
    }
    __syncthreads();

    const float e1i = e1[row];
    const unsigned int* adjRow =
        (const unsigned int*)(adj + (size_t)row * NCOL);   // 4 bools / load

    // ---- Pass 1: masked row max -------------------------------------------
    float m = -INFINITY;
    for (int j4 = tid; j4 < NCOL / 4; j4 += 256) {
        unsigned int a4 = adjRow[j4];
        if (!a4) continue;                       // ~81% of words at 5% density
        const int jb = j4 * 4;
#pragma unroll
        for (int cc = 0; cc < 4; ++cc) {
            if ((a4 >> (8 * cc)) & 0xffu) {
                float s = leaky(e1i + s_e2[jb + cc]);
                m = fmaxf(m, s);
            }
        }
    }
#pragma unroll
    for (int off = 16; off > 0; off >>= 1)
        m = fmaxf(m, __shfl_xor(m, off, 32));
    if ((tid & 31) == 0) s_red[tid >> 5] = m;
    __syncthreads();
    const float M = fmaxf(fmaxf(fmaxf(s_red[0], s_red[1]), fmaxf(s_red[2], s_red[3])),
                          fmaxf(fmaxf(s_red[4], s_red[5]), fmaxf(s_red[6], s_red[7])));
    __syncthreads();                              // before s_red reuse

    // ---- Pass 2: masked exp-sum -------------------------------------------
    float l = 0.0f;
    for (int j4 = tid; j4 < NCOL / 4; j4 += 256) {
        unsigned int a4 = adjRow[j4];
        if (!a4) continue;
        const int jb = j4 * 4;
#pragma unroll
        for (int cc = 0; cc < 4; ++cc) {
            if ((a4 >> (8 * cc)) & 0xffu) {
                float s = leaky(e1i + s_e2[jb + cc]);
                l += __expf(s - M);
            }
        }
    }
#pragma unroll
    for (int off = 16; off > 0; off >>= 1)
        l += __shfl_xor(l, off, 32);
    if ((tid & 31) == 0) s_red[tid >> 5] = l;
    __syncthreads();
    const float L = ((s_red[0] + s_red[1]) + (s_red[2] + s_red[3])) +
                    ((s_red[4] + s_red[5]) + (s_red[6] + s_red[7]));
    const float invL = 1.0f / L;                  // diagonal guarantees L >= exp(0)

    // ---- Pass 3: write normalized row (NT b128 stores) --------------------
    v4f* outRow = (v4f*)(out + (size_t)row * NCOL);
    for (int j4 = tid; j4 < NCOL / 4; j4 += 256) {
        unsigned int a4 = adjRow[j4];
        float o0 = 0.0f, o1 = 0.0f, o2 = 0.0f, o3 = 0.0f;
        if (a4) {
            const int jb = j4 * 4;
            if (a4 & 0x000000ffu) o0 = __expf(leaky(e1i + s_e2[jb + 0]) - M) * invL;
            if (a4 & 0x0000ff00u) o1 = __expf(leaky(e1i + s_e2[jb + 1]) - M) * invL;
            if (a4 & 0x00ff0000u) o2 = __expf(leaky(e1i + s_e2[jb + 2]) - M) * invL;
            if (a4 & 0xff000000u) o3 = __expf(leaky(e1i + s_e2[jb + 3]) - M) * invL;
        }
        v4f o = { o0, o1, o2, o3 };
        __builtin_nontemporal_store(o, &outRow[j4]);
    }
}

// ---------------------------------------------------------------------------
extern "C" void kernel_launch(void* const* d_in, const int* in_sizes, int n_in,
                              void* d_out, int out_size, void* d_ws, size_t ws_size,
                              hipStream_t stream)
{
    const float*         h   = (const float*)d_in[0];          // (8192, 256) f32
    const unsigned char* adj = (const unsigned char*)d_in[1];  // (8192, 8192) bool
    const float*         a   = (const float*)d_in[2];          // (256, 2) f32

    float* e1 = (float*)d_ws;          // 8192 f32
    float* e2 = e1 + NROW;             // 8192 f32  (64 KB total scratch)

    gat_e_wmma_kernel<<<NROW / 16, 32, 0, stream>>>(h, a, e1, e2);
    gat_softmax_kernel<<<NROW, 256, 0, stream>>>(adj, e1, e2, (float*)d_out);
}